// FeatureExtractor_9208409883056
// MI455X (gfx1250) — compile-verified
//
#include <hip/hip_runtime.h>
#include <math.h>

// ---------------------------------------------------------------------------
// FeatureExtractor for MI455X (gfx1250, wave32)
//   out[b*10 + {0..9}] = [mx, mx-mn, var, std, fftm, PA, A2, PH, A1, PWHH]
//   Kernel 1: streaming stats/peak features (memory-bound, LDS-staged)
//   Kernel 2: |FFT|.mean() via four-step DFT (8192 = 64 x 128) with
//             V_WMMA_F32_16X16X4_F32; twiddle fragments generated by
//             incremental complex rotation (3 sincos per tile, 4 FMA/step);
//             signal staged with GLOBAL_LOAD_ASYNC_TO_LDS_B32.
// ---------------------------------------------------------------------------

typedef __attribute__((ext_vector_type(2))) float v2f;
typedef __attribute__((ext_vector_type(8))) float v8f;

#define SIG_L   8192
#define FFT_N1  64
#define FFT_N2  128
#define LDSP    129          // row pitch (odd -> conflict-free column walks)
#define SRF     30.0f

__device__ __forceinline__ v8f wmma_f32(v2f a, v2f b, v8f c) {
  // D = A(16x4) * B(4x16) + C(16x16), fp32.  NEG/NEG_HI must be 0 for f32.
  return __builtin_amdgcn_wmma_f32_16x16x4_f32(
      /*neg_a=*/false, a, /*neg_b=*/false, b,
      /*c_mod=*/(short)0, c, /*reuse_a=*/false, /*reuse_b=*/false);
}

// ---------------------------------------------------------------------------
// Kernel 1: scalar / scan features. One 256-thread block per signal.
// ---------------------------------------------------------------------------
__global__ __launch_bounds__(256) void stats_kernel(
    const float* __restrict__ x, float* __restrict__ out) {
  __shared__ float s[SIG_L];        // 32 KB signal stage
  __shared__ float rf[256 * 4];     // float reduction scratch
  __shared__ int   rixx[256 * 5];   // int reduction scratch

  const int b   = blockIdx.x;
  const int tid = threadIdx.x;
  const float* sig = x + (size_t)b * SIG_L;

  for (int i = tid; i < SIG_L; i += 256) {
    if (i + 2048 < SIG_L) __builtin_prefetch(sig + i + 2048, 0, 0);
    s[i] = sig[i];
  }
  __syncthreads();

  // ---- pass 1: max/min/sum/sumsq + peak/valley stats -----------------------
  float mx = -__builtin_inff(), mn = __builtin_inff();
  float sum = 0.f, sumsq = 0.f;
  int npk = 0, nvl = 0, p0 = SIG_L, v0 = SIG_L, vlast = -1;
  for (int i = tid; i < SIG_L; i += 256) {
    float v = s[i];
    mx = fmaxf(mx, v); mn = fminf(mn, v);
    sum += v; sumsq += v * v;
    if (i >= 1 && i <= SIG_L - 2) {
      float a = s[i - 1], c = s[i + 1];
      if (v > a && v > c) { npk++; p0 = min(p0, i); }
      if (v < a && v < c) { nvl++; v0 = min(v0, i); vlast = max(vlast, i); }
    }
  }
  rf[tid] = mx; rf[256 + tid] = mn; rf[512 + tid] = sum; rf[768 + tid] = sumsq;
  rixx[tid] = npk; rixx[256 + tid] = nvl; rixx[512 + tid] = p0;
  rixx[768 + tid] = v0; rixx[1024 + tid] = vlast;
  __syncthreads();
  for (int off = 128; off > 0; off >>= 1) {
    if (tid < off) {
      rf[tid]        = fmaxf(rf[tid], rf[tid + off]);
      rf[256 + tid]  = fminf(rf[256 + tid], rf[256 + tid + off]);
      rf[512 + tid]  += rf[512 + tid + off];
      rf[768 + tid]  += rf[768 + tid + off];
      rixx[tid]        += rixx[tid + off];
      rixx[256 + tid]  += rixx[256 + tid + off];
      rixx[512 + tid]  = min(rixx[512 + tid],  rixx[512 + tid + off]);
      rixx[768 + tid]  = min(rixx[768 + tid],  rixx[768 + tid + off]);
      rixx[1024 + tid] = max(rixx[1024 + tid], rixx[1024 + tid + off]);
    }
    __syncthreads();
  }
  const float MX = rf[0], MN = rf[256], SUM = rf[512], SQ = rf[768];
  const int NPK = rixx[0], NVL = rixx[256];
  const int P0 = rixx[512], V0 = rixx[768], VLAST = rixx[1024];
  __syncthreads();

  // ---- pass 2a: second valley (min valley index strictly after V0) --------
  int v1 = SIG_L;
  for (int i = tid; i < SIG_L; i += 256) {
    if (i >= 1 && i <= SIG_L - 2) {
      float v = s[i];
      if (v < s[i - 1] && v < s[i + 1] && i > V0) v1 = min(v1, i);
    }
  }
  rixx[tid] = v1;
  __syncthreads();
  for (int off = 128; off > 0; off >>= 1) {
    if (tid < off) rixx[tid] = min(rixx[tid], rixx[tid + off]);
    __syncthreads();
  }
  const int V1 = rixx[0];
  __syncthreads();

  // ---- pass 2b: trapezoid integrals + half-height width --------------------
  const int p0c = min(P0, SIG_L - 1), v0c = min(V0, SIG_L - 1);
  const int v1c = min(V1, SIG_L - 1), vlc = max(VLAST, 0);
  const float hh = (s[p0c] + s[v0c]) * 0.5f;   // == sig[v0] + PH * 0.5
  float PA = 0.f, A2 = 0.f, A1 = 0.f;
  int li = SIG_L, riq = -1;
  for (int i = tid; i < SIG_L; i += 256) {
    float v = s[i];
    if (i <= SIG_L - 2) {
      float seg = (s[i] + s[i + 1]) * (0.5f / SRF);
      if (i >= v0c && i <= vlc - 2) PA += seg;
      if (i >= p0c && i <= v1c - 2) A2 += seg;
      if (i >= v0c && i <= p0c - 2) A1 += seg;
    }
    if (i >= v0c && i <  p0c && v >= hh) li  = min(li, i);
    if (i >  v0c && i <= p0c && v >= hh) riq = max(riq, i);
  }
  rf[tid] = PA; rf[256 + tid] = A2; rf[512 + tid] = A1;
  rixx[tid] = li; rixx[256 + tid] = riq;
  __syncthreads();
  for (int off = 128; off > 0; off >>= 1) {
    if (tid < off) {
      rf[tid]       += rf[tid + off];
      rf[256 + tid] += rf[256 + tid + off];
      rf[512 + tid] += rf[512 + tid + off];
      rixx[tid]       = min(rixx[tid], rixx[tid + off]);
      rixx[256 + tid] = max(rixx[256 + tid], rixx[256 + tid + off]);
    }
    __syncthreads();
  }

  if (tid == 0) {
    const float n   = (float)SIG_L;
    const float var = (SQ - SUM * SUM / n) / (n - 1.0f);
    float* o = out + (size_t)b * 10;
    o[0] = MX; o[1] = MX - MN; o[2] = var; o[3] = sqrtf(var);
    const bool valid = (NPK >= 1) && (NVL >= 2);
    if (valid) {
      const float PH  = s[p0c] - s[v0c];
      const int   liF = (rixx[0]   == SIG_L) ? v0c : rixx[0];
      const int   riF = (rixx[256] == -1)    ? p0c : rixx[256];
      o[5] = rf[0];            // PA
      o[6] = rf[256];          // A2
      o[7] = PH;               // PH
      o[8] = rf[512];          // A1
      o[9] = (float)(riF - liF) / SRF;  // PWHH
    } else {
      o[5] = 0.f; o[6] = 0.f; o[7] = 0.f; o[8] = 0.f; o[9] = 0.f;
    }
  }
}

// ---------------------------------------------------------------------------
// Kernel 2: fftm = mean(|DFT_8192(sig)|) via four-step DFT with fp32 WMMA.
//   n = n1 + 64*n2 ; k = 128*k1 + k2
//   Stage1: Y[n1][k2]  = sum_n2 D[n1][n2] * e^{-2pi i n2 k2 / 128}
//   Twiddle: Y' = Y * e^{-2pi i n1 k2 / 8192}
//   Stage2: X[k1][k2]  = sum_n1 e^{-2pi i n1 k1 / 64} * Y'[n1][k2]
//   fftm = sum |X| / 8192  (order-free -> no reordering needed)
// One 256-thread block (8 wave32) per signal; 32 16x16 tiles per stage,
// 4 tiles per wave. Twiddle fragments advance by complex rotation
// (w^(k+4) = w^k * w^4) instead of per-step sincos.
// ---------------------------------------------------------------------------
__global__ __launch_bounds__(256) void fft_mag_mean_kernel(
    const float* __restrict__ x, float* __restrict__ out) {
  extern __shared__ float lds[];
  const int D_OFF   = 0;
  const int YRE_OFF = FFT_N1 * LDSP;
  const int YIM_OFF = 2 * FFT_N1 * LDSP;
  const int RED_OFF = 3 * FFT_N1 * LDSP;

  const int b   = blockIdx.x;
  const int tid = threadIdx.x;
  const float* sig = x + (size_t)b * SIG_L;

  // Async scatter stage-in: D[n1][n2] = sig[n1 + 64*n2]. Each lane supplies
  // its own LDS destination; tracked by ASYNCcnt.
  {
    const unsigned dynbase = __builtin_amdgcn_groupstaticsize();
    for (int i = tid; i < SIG_L; i += 256) {
      unsigned ldsOff = dynbase +
          (unsigned)((D_OFF + (i & 63) * LDSP + (i >> 6)) * sizeof(float));
      unsigned long long ga = (unsigned long long)(const void*)(sig + i);
      asm volatile("global_load_async_to_lds_b32 %0, %1, off"
                   :: "v"(ldsOff), "v"(ga) : "memory");
    }
    asm volatile("s_wait_asynccnt 0x0" ::: "memory");
  }
  __syncthreads();

  const int wave = tid >> 5;
  const int lane = tid & 31;
  const int l16  = lane & 15;
  const int half = (lane >> 4) & 1;
  const float W128 = -6.28318530717958647692f / 128.0f;
  const float W64  = -6.28318530717958647692f / 64.0f;
  const float WN   = -6.28318530717958647692f / 8192.0f;

  // ---- Stage 1: M=n1(64) x N=k2(128), K=n2(128). A=D tile from LDS,
  //      B = DFT128 fragment; lane's B entries are w^kk with w=e^{-2pi i ncol/128}
  for (int t = wave * 4; t < wave * 4 + 4; ++t) {
    const int m0 = (t >> 3) * 16, n0 = (t & 7) * 16;
    const int ncol = n0 + l16;
    const int arow = D_OFF + (m0 + l16) * LDSP;
    // init phases at kk = 2*half and 2*half+1; step factor w^4
    float s0, c0, s1, c1, ss, cs;
    __sincosf((float)(((2 * half)     * ncol) & 127) * W128, &s0, &c0);
    __sincosf((float)(((2 * half + 1) * ncol) & 127) * W128, &s1, &c1);
    __sincosf((float)((4 * ncol) & 127) * W128, &ss, &cs);
    v8f cre = {}; v8f cim = {};
    for (int k = 0; k < FFT_N2; k += 4) {
      const int kk = k + 2 * half;
      v2f a  = { lds[arow + kk], lds[arow + kk + 1] };
      v2f bc = { c0, c1 };
      v2f bs = { s0, s1 };                 // already -sin (negative angle)
      cre = wmma_f32(a, bc, cre);          // Yre += D * cos
      cim = wmma_f32(a, bs, cim);          // Yim += D * (-sin)
      // rotate both phase sequences by w^4
      const float nc0 = c0 * cs - s0 * ss, ns0 = s0 * cs + c0 * ss;
      const float nc1 = c1 * cs - s1 * ss, ns1 = s1 * cs + c1 * ss;
      c0 = nc0; s0 = ns0; c1 = nc1; s1 = ns1;
    }
    for (int v = 0; v < 8; ++v) {
      const int m = m0 + v + 8 * half;
      lds[YRE_OFF + m * LDSP + ncol] = cre[v];
      lds[YIM_OFF + m * LDSP + ncol] = cim[v];
    }
  }
  __syncthreads();

  // ---- Twiddle: Y' = Y * e^{-2pi i n1 k2 / 8192} (one-off, exact int phase)
  for (int i = tid; i < SIG_L; i += 256) {
    const int n1 = i >> 7, k2 = i & 127;
    const int idx = (n1 * k2) & (SIG_L - 1);
    float sn, csn;
    __sincosf((float)idx * WN, &sn, &csn);      // (cos, -sin) of +angle
    const int a = n1 * LDSP + k2;
    const float re = lds[YRE_OFF + a], im = lds[YIM_OFF + a];
    lds[YRE_OFF + a] = re * csn - im * sn;
    lds[YIM_OFF + a] = re * sn + im * csn;
  }
  __syncthreads();

  // ---- Stage 2: M=k1(64) x N=k2(128), K=n1(64). A = DFT64 fragment
  //      (w^kk, w=e^{-2pi i mrow/64}), B = Y' tile from LDS.
  float partial = 0.f;
  for (int t = wave * 4; t < wave * 4 + 4; ++t) {
    const int m0 = (t >> 3) * 16, n0 = (t & 7) * 16;
    const int ncol = n0 + l16;
    const int mrow = m0 + l16;
    float s0, c0, s1, c1, ss, cs;
    __sincosf((float)(((2 * half)     * mrow) & 63) * W64, &s0, &c0);
    __sincosf((float)(((2 * half + 1) * mrow) & 63) * W64, &s1, &c1);
    __sincosf((float)((4 * mrow) & 63) * W64, &ss, &cs);
    v8f xre = {}; v8f xim = {};
    for (int k = 0; k < FFT_N1; k += 4) {
      const int kk = k + 2 * half;
      v2f ac  = { c0, c1 };
      v2f as  = { s0, s1 };       // -sin
      v2f ams = { -s0, -s1 };     // +sin (f32 WMMA has no A-neg modifier)
      v2f bre = { lds[YRE_OFF + kk * LDSP + ncol],
                  lds[YRE_OFF + (kk + 1) * LDSP + ncol] };
      v2f bim = { lds[YIM_OFF + kk * LDSP + ncol],
                  lds[YIM_OFF + (kk + 1) * LDSP + ncol] };
      // (c - i sin)(Yre + i Yim): Xre = c*Yre + sin*Yim ; Xim = c*Yim - sin*Yre
      xre = wmma_f32(ac,  bre, xre);
      xre = wmma_f32(ams, bim, xre);
      xim = wmma_f32(ac,  bim, xim);
      xim = wmma_f32(as,  bre, xim);
      const float nc0 = c0 * cs - s0 * ss, ns0 = s0 * cs + c0 * ss;
      const float nc1 = c1 * cs - s1 * ss, ns1 = s1 * cs + c1 * ss;
      c0 = nc0; s0 = ns0; c1 = nc1; s1 = ns1;
    }
    for (int v = 0; v < 8; ++v)
      partial += sqrtf(xre[v] * xre[v] + xim[v] * xim[v]);
  }

  // deterministic block reduction of |X| partial sums
  float* red = lds + RED_OFF;
  red[tid] = partial;
  __syncthreads();
  for (int off = 128; off > 0; off >>= 1) {
    if (tid < off) red[tid] += red[tid + off];
    __syncthreads();
  }
  if (tid == 0) out[(size_t)b * 10 + 4] = red[0] * (1.0f / (float)SIG_L);
}

// ---------------------------------------------------------------------------
extern "C" void kernel_launch(void* const* d_in, const int* in_sizes, int n_in,
                              void* d_out, int out_size, void* d_ws, size_t ws_size,
                              hipStream_t stream) {
  const float* x = (const float*)d_in[0];
  float* out = (float*)d_out;
  const int nb = in_sizes[0] / SIG_L;   // 2048 signals

  stats_kernel<<<dim3(nb), dim3(256), 0, stream>>>(x, out);

  const size_t dyn_lds = (size_t)(3 * FFT_N1 * LDSP + 256) * sizeof(float);
  fft_mag_mean_kernel<<<dim3(nb), dim3(256), dyn_lds, stream>>>(x, out);
}